// LocalLasso_78718160601453
// MI455X (gfx1250) — compile-verified
//
#include <hip/hip_runtime.h>

typedef float v2f __attribute__((ext_vector_type(2)));
typedef float v4f __attribute__((ext_vector_type(4)));
typedef float v8f __attribute__((ext_vector_type(8)));

#define S_DIM 8192
#define C_DIM 64
#define B_DIM 256
#define BCHUNK 8            // batches per wave (software-pipelined)
#define WAVES_PER_BLOCK 8   // 256 threads, wave32

// Load one 16-row x-tile (this lane's slice, WMMA A-layout) into registers.
__device__ __forceinline__ void load_tile(v2f (&dst)[16], const float* __restrict__ xRow) {
#pragma unroll
  for (int j = 0; j < 16; ++j)
    dst[j] = __builtin_nontemporal_load((const v2f*)(xRow + 4 * j));
}

// p = x*W elementwise, then row-sum via V_WMMA_F32_16X16X4_F32 with all-ones B.
__device__ __forceinline__ v8f compute_tile(const v2f (&xv)[16], const v2f (&wv)[16],
                                            v2f ones) {
  v8f acc0 = {0.f, 0.f, 0.f, 0.f, 0.f, 0.f, 0.f, 0.f};
  v8f acc1 = {0.f, 0.f, 0.f, 0.f, 0.f, 0.f, 0.f, 0.f};
#pragma unroll
  for (int j = 0; j < 16; j += 2) {
    v2f p0 = xv[j] * wv[j];           // v_pk_mul_f32
    v2f p1 = xv[j + 1] * wv[j + 1];
    acc0 = __builtin_amdgcn_wmma_f32_16x16x4_f32(
        false, p0, false, ones, (short)0, acc0, false, false);
    acc1 = __builtin_amdgcn_wmma_f32_16x16x4_f32(
        false, p1, false, ones, (short)0, acc1, false, false);
  }
  return acc0 + acc1;
}

// out[b, s] = sum_c x[b,s,c] * W[s,c]
__global__ __launch_bounds__(256) void LocalLasso_wmma_kernel(
    const float* __restrict__ x, const float* __restrict__ W,
    float* __restrict__ out)
{
  const int lane  = threadIdx.x & 31;      // wave32
  const int wave  = threadIdx.x >> 5;      // 0..7
  const int sTile = blockIdx.x * WAVES_PER_BLOCK + wave;   // 0..511
  const int s0    = sTile * 16;
  const int r     = lane & 15;             // matrix row (s0 + r)
  const int h     = lane >> 4;             // K half: 0 -> K=0,1 ; 1 -> K=2,3

  // Preload W tile into registers in WMMA A-matrix layout (temporal: L2-resident).
  const float* wRow = W + (size_t)(s0 + r) * C_DIM + 2 * h;
  v2f wv[16];
#pragma unroll
  for (int j = 0; j < 16; ++j)
    wv[j] = *(const v2f*)(wRow + 4 * j);

  const v2f ones = {1.0f, 1.0f};           // all-ones B matrix (layout-free)
  const int b0 = blockIdx.y * BCHUNK;

  const size_t bStride = (size_t)S_DIM * C_DIM;          // elements per batch
  const float* xRow0 = x + ((size_t)b0 * S_DIM + s0 + r) * C_DIM + 2 * h;
  const char*  pfBase = (const char*)(x + ((size_t)b0 * S_DIM + s0) * C_DIM)
                        + (size_t)lane * 128;            // 32 lanes cover 4KB tile

  // Software pipeline: tile bi+1 streams into the alternate register buffer
  // while tile bi runs its mul+WMMA chain -> 16 loads (4KB/wave) always in flight.
  v2f buf[2][16];
  load_tile(buf[0], xRow0);

#pragma unroll
  for (int bi = 0; bi < BCHUNK; ++bi) {
    if (bi + 1 < BCHUNK)
      load_tile(buf[(bi + 1) & 1], xRow0 + (size_t)(bi + 1) * bStride);
    if (bi + 2 < BCHUNK)   // pull tile bi+2 toward WGP caches (global_prefetch_b8)
      __builtin_prefetch(pfBase + (size_t)(bi + 2) * bStride * sizeof(float), 0, 3);

    v8f acc = compute_tile(buf[bi & 1], wv, ones);

    // All D columns identical: lane 0 holds rows 0..7, lane 16 holds rows 8..15.
    if (r == 0) {
      float* o = out + (size_t)(b0 + bi) * S_DIM + s0 + h * 8;
      v4f lo = {acc[0], acc[1], acc[2], acc[3]};
      v4f hi = {acc[4], acc[5], acc[6], acc[7]};
      __builtin_nontemporal_store(lo, (v4f*)o);
      __builtin_nontemporal_store(hi, (v4f*)(o + 4));
    }
  }
}

extern "C" void kernel_launch(void* const* d_in, const int* in_sizes, int n_in,
                              void* d_out, int out_size, void* d_ws, size_t ws_size,
                              hipStream_t stream) {
  const float* x = (const float*)d_in[0];   // [256, 8192, 64] fp32
  const float* W = (const float*)d_in[1];   // [8192, 64] fp32
  float* out = (float*)d_out;               // [256, 8192, 1] fp32

  dim3 block(32 * WAVES_PER_BLOCK, 1, 1);                       // 256 thr = 8 waves
  dim3 grid(S_DIM / 16 / WAVES_PER_BLOCK, B_DIM / BCHUNK, 1);   // (64, 32)
  hipLaunchKernelGGL(LocalLasso_wmma_kernel, grid, block, 0, stream, x, W, out);
}